// GAEEncoder_59931973648925
// MI455X (gfx1250) — compile-verified
//
#include <hip/hip_runtime.h>

typedef float v2f __attribute__((ext_vector_type(2)));
typedef float v8f __attribute__((ext_vector_type(8)));

static constexpr int NN  = 100000;   // nodes
static constexpr int NE  = 1600000;  // edges
static constexpr int KIN = 256;      // F_IN
static constexpr int D1  = 32;       // conv1 out
static constexpr int D2  = 16;       // conv2 out

static inline int cdiv(long long a, long long b) { return (int)((a + b - 1) / b); }

// ---------------- small elementwise kernels ----------------

__global__ void k_fill_one(float* __restrict__ p, int n) {
  int i = blockIdx.x * blockDim.x + threadIdx.x;
  if (i < n) p[i] = 1.0f;
}

__global__ void k_deg_accum(const long long* __restrict__ col, const float* __restrict__ w,
                            float* __restrict__ deg, int E) {
  int e = blockIdx.x * blockDim.x + threadIdx.x;
  if (e < E) atomicAdd(&deg[(int)col[e]], w[e]);
}

__global__ void k_rsqrt_inplace(float* __restrict__ d, int n) {
  int i = blockIdx.x * blockDim.x + threadIdx.x;
  if (i < n) { float v = d[i]; d[i] = (v > 0.0f) ? rsqrtf(v) : 0.0f; }
}

__global__ void k_edge_norm(const long long* __restrict__ row, const long long* __restrict__ col,
                            const float* __restrict__ w, const float* __restrict__ dinv,
                            float* __restrict__ norm, int E, int N) {
  int e = blockIdx.x * blockDim.x + threadIdx.x;
  if (e < E) {
    norm[e] = dinv[(int)row[e]] * w[e] * dinv[(int)col[e]];
  } else if (e < E + N) {
    float dv = dinv[e - E];          // self loop: weight 1.0
    norm[e] = dv * dv;
  }
}

// buf laid out [n, d] row-major; fill each row with bias vector b[d]
__global__ void k_fill_bias(float* __restrict__ buf, const float* __restrict__ b,
                            int total, int d) {
  int i = blockIdx.x * blockDim.x + threadIdx.x;
  if (i < total) buf[i] = b[i % d];
}

// ---------------- f32 WMMA GEMM: C[nrows, NT*16] = op(A[nrows,K]) @ W[K, NT*16] ----------------
// One wave computes a 16 x (NT*16) tile using V_WMMA_F32_16X16X4_F32.
// A layout (ISA 7.12.2, 32-bit A 16x4): lane l holds row (l&15); K-pair {2*(l>>4), 2*(l>>4)+1}
// in VGPR0/VGPR1 of the fragment. B mirrored: N = (l&15) striped across lanes,
// K-pair selected by (l>>4).
// STREAM_A: A is a read-once stream (x) -> non-temporal loads to keep L2 for the
// graph-phase working set (edge index / norm / feature buffers all have reuse).
template<int NT, bool RELU_IN, bool STREAM_A, int K>
__global__ void k_gemm_wmma(const float* __restrict__ A, const float* __restrict__ W,
                            float* __restrict__ C, int nrows) {
  const int NCOL = NT * 16;
  int wave = (int)((blockIdx.x * (long long)blockDim.x + threadIdx.x) >> 5);
  int row0 = wave * 16;
  if (row0 >= nrows) return;             // whole wave exits together: EXEC stays all-1s for WMMA

  int l    = threadIdx.x & 31;
  int half = l >> 4;
  int lm   = l & 15;

  v8f acc[NT];
#pragma unroll
  for (int t = 0; t < NT; ++t) acc[t] = {};

  const float* __restrict__ arow = A + (size_t)(row0 + lm) * K;

  auto step = [&](int k0) {
    int ka = k0 + 2 * half;
    float a0, a1;
    if (STREAM_A) {
      a0 = __builtin_nontemporal_load(arow + ka);
      a1 = __builtin_nontemporal_load(arow + ka + 1);
    } else {
      a0 = arow[ka];
      a1 = arow[ka + 1];
    }
    if (RELU_IN) { a0 = fmaxf(a0, 0.0f); a1 = fmaxf(a1, 0.0f); }
    v2f a; a.x = a0; a.y = a1;

#pragma unroll
    for (int t = 0; t < NT; ++t) {
      v2f b;
      b.x = W[(size_t)ka * NCOL + t * 16 + lm];
      b.y = W[(size_t)(ka + 1) * NCOL + t * 16 + lm];
      acc[t] = __builtin_amdgcn_wmma_f32_16x16x4_f32(
          /*neg_a=*/false, a, /*neg_b=*/false, b,
          /*c_mod=*/(short)0, acc[t], /*reuse_a=*/false, /*reuse_b=*/false);
    }
  };

  if constexpr (K <= 32) {
    // small-K GEMM: fully unrolled, B fragments hoisted to registers by the compiler
#pragma unroll
    for (int k0 = 0; k0 < K; k0 += 4) step(k0);
  } else {
#pragma unroll 8
    for (int k0 = 0; k0 < K; k0 += 4) {
      if ((k0 & 127) == 0) __builtin_prefetch(arow + k0 + 128, 0, 0);  // global_prefetch_b8
      step(k0);
    }
  }

  // 32-bit C/D layout: VGPR g, lanes 0-15 -> M=g, lanes 16-31 -> M=g+8, N=(l&15)
#pragma unroll
  for (int t = 0; t < NT; ++t) {
#pragma unroll
    for (int g = 0; g < 8; ++g) {
      C[(size_t)(row0 + g + 8 * half) * NCOL + t * 16 + lm] = acc[t][g];
    }
  }
}

// ---------------- edge scatter-add: dst[col] += norm * src[row], d channels ----------------
// d4 = d/4 threads per edge, each handling a float4 slice; self loops are edges [E, E+N).
__global__ void k_scatter(const long long* __restrict__ row, const long long* __restrict__ col,
                          const float* __restrict__ norm, const float* __restrict__ src,
                          float* __restrict__ dst, int E, int N, int d4, int d) {
  long long tid = blockIdx.x * (long long)blockDim.x + threadIdx.x;
  long long total = (long long)(E + N) * d4;
  if (tid >= total) return;
  int e = (int)(tid / d4);
  int c = (int)(tid % d4) * 4;

  int r, cd;
  if (e < E) { r = (int)row[e]; cd = (int)col[e]; }
  else       { r = cd = e - E; }

  float nv = norm[e];
  const float4 v = *(const float4*)(src + (size_t)r * d + c);
  float* o = dst + (size_t)cd * d + c;
  atomicAdd(o + 0, nv * v.x);
  atomicAdd(o + 1, nv * v.y);
  atomicAdd(o + 2, nv * v.z);
  atomicAdd(o + 3, nv * v.w);
}

// ---------------- launch ----------------

extern "C" void kernel_launch(void* const* d_in, const int* in_sizes, int n_in,
                              void* d_out, int out_size, void* d_ws, size_t ws_size,
                              hipStream_t stream) {
  const float*     x  = (const float*)d_in[0];       // [NN, KIN]
  const long long* ei = (const long long*)d_in[1];   // [2, NE] int64: rows then cols
  const float*     w  = (const float*)d_in[2];       // [NE]
  const float*     W1 = (const float*)d_in[3];       // [KIN, D1]
  const float*     b1 = (const float*)d_in[4];       // [D1]
  const float*     W2 = (const float*)d_in[5];       // [D1, D2]
  const float*     b2 = (const float*)d_in[6];       // [D2]
  float*           out = (float*)d_out;              // [NN, D2]

  const long long* erow = ei;
  const long long* ecol = ei + NE;

  // workspace carve-up (floats): dinv[N] | norm[E+N] | hp1[N*D1] | agg1[N*D1] | hp2[N*D2]
  float* f    = (float*)d_ws;
  float* dinv = f;
  float* norm = dinv + NN;
  float* hp1  = norm + (NE + NN);
  float* agg1 = hp1 + (size_t)NN * D1;
  float* hp2  = agg1 + (size_t)NN * D1;

  const int B = 256;

  // gcn_norm
  k_fill_one<<<cdiv(NN, B), B, 0, stream>>>(dinv, NN);                       // deg = 1 (self loop)
  k_deg_accum<<<cdiv(NE, B), B, 0, stream>>>(ecol, w, dinv, NE);             // deg += w by col
  k_rsqrt_inplace<<<cdiv(NN, B), B, 0, stream>>>(dinv, NN);                  // dinv
  k_edge_norm<<<cdiv(NE + NN, B), B, 0, stream>>>(erow, ecol, w, dinv, norm, NE, NN);

  // conv1: hp1 = x @ W1 (WMMA f32, x streamed non-temporally),
  //        agg1 = b1 + scatter(norm * hp1[row])
  k_gemm_wmma<2, false, true, KIN>
      <<<cdiv((long long)(NN / 16) * 32, B), B, 0, stream>>>(x, W1, hp1, NN);
  k_fill_bias<<<cdiv((long long)NN * D1, B), B, 0, stream>>>(agg1, b1, NN * D1, D1);
  k_scatter<<<cdiv((long long)(NE + NN) * (D1 / 4), B), B, 0, stream>>>(
      erow, ecol, norm, hp1, agg1, NE, NN, D1 / 4, D1);

  // conv2: hp2 = relu(agg1) @ W2 (WMMA f32, ReLU fused into A-load, fully unrolled K=32),
  //        out = b2 + scatter
  k_gemm_wmma<1, true, false, D1>
      <<<cdiv((long long)(NN / 16) * 32, B), B, 0, stream>>>(agg1, W2, hp2, NN);
  k_fill_bias<<<cdiv((long long)NN * D2, B), B, 0, stream>>>(out, b2, NN * D2, D2);
  k_scatter<<<cdiv((long long)(NE + NN) * (D2 / 4), B), B, 0, stream>>>(
      erow, ecol, norm, hp2, out, NE, NN, D2 / 4, D2);
}